// CircuitGNN_59596966199647
// MI455X (gfx1250) — compile-verified
//
#include <hip/hip_runtime.h>

typedef float v2f __attribute__((ext_vector_type(2)));
typedef float v8f __attribute__((ext_vector_type(8)));

#define HIDDEN 64

// ---------------- utility kernels ----------------

__global__ void k_fill(float* __restrict__ p, float v, int n) {
  int i = blockIdx.x * blockDim.x + threadIdx.x;
  if (i < n) p[i] = v;
}

__global__ void k_degree(const int* __restrict__ dst, float* __restrict__ deg, int E) {
  int i = blockIdx.x * blockDim.x + threadIdx.x;
  if (i < E) atomicAdd(&deg[dst[i]], 1.0f);
}

__global__ void k_rsqrt(float* __restrict__ deg, int n) {
  int i = blockIdx.x * blockDim.x + threadIdx.x;
  if (i < n) deg[i] = rsqrtf(fmaxf(deg[i], 1.0f));
}

// ---------------- WMMA GEMM: C[M,64] = A[M,K] @ W[K,64] (fp32) ----------------
// W is staged in LDS once per block in a paired-K layout:
//   Wl[(k>>1)*128 + col*2 + (k&1)]  ->  B fragment (W[ka][col], W[ka+1][col])
//   is one ds_load_b64 (ka is always even: ka = k4 + half*2).
// A layout (ISA 7.12.2, 32-bit A 16x4):
//   lanes 0-15 : row M=lane,    v0=K+0, v1=K+1
//   lanes 16-31: row M=lane-16, v0=K+2, v1=K+3   -> one global_load_b64 per step.
// C/D: 8 VGPRs; lanes 0-15 -> M=j, lanes 16-31 -> M=j+8, N=lane%16.
template <int K>
__global__ void k_gemm_wmma(const float* __restrict__ A, const float* __restrict__ W,
                            float* __restrict__ C, int M) {
  __shared__ float Wl[K * HIDDEN];

  // cooperative staging (ALL waves participate, then barrier)
  for (int i = threadIdx.x; i < K * HIDDEN; i += blockDim.x) {
    int k = i >> 6;      // row of W
    int c = i & 63;      // col of W
    Wl[(k >> 1) * 128 + c * 2 + (k & 1)] = W[i];
  }
  __syncthreads();

  int wave = blockIdx.x * (blockDim.x >> 5) + (threadIdx.x >> 5);
  int rowBase = wave * 16;
  if (rowBase < M) {                        // wave-uniform: EXEC all-ones inside
    int lane = threadIdx.x & 31;
    int half = lane >> 4;                   // 0 or 1
    int lm   = lane & 15;

    int arow = rowBase + lm;
    if (arow >= M) arow = M - 1;            // clamp (M is a multiple of 16 here)
    const float* Arow = A + (size_t)arow * K;

    v8f acc0 = {}, acc1 = {}, acc2 = {}, acc3 = {};

    for (int k4 = 0; k4 < K; k4 += 4) {
      int ka = k4 + half * 2;               // even
      v2f a = *(const v2f*)(Arow + ka);     // global_load_b64
      const float* Bl = Wl + (ka >> 1) * 128;
      v2f b0 = *(const v2f*)(Bl + 2 * lm);          // ds_load_b64
      v2f b1 = *(const v2f*)(Bl + 2 * (16 + lm));
      v2f b2 = *(const v2f*)(Bl + 2 * (32 + lm));
      v2f b3 = *(const v2f*)(Bl + 2 * (48 + lm));
      acc0 = __builtin_amdgcn_wmma_f32_16x16x4_f32(false, a, false, b0, (short)0, acc0, false, false);
      acc1 = __builtin_amdgcn_wmma_f32_16x16x4_f32(false, a, false, b1, (short)0, acc1, false, false);
      acc2 = __builtin_amdgcn_wmma_f32_16x16x4_f32(false, a, false, b2, (short)0, acc2, false, false);
      acc3 = __builtin_amdgcn_wmma_f32_16x16x4_f32(false, a, false, b3, (short)0, acc3, false, false);
    }

    int orow = rowBase + half * 8;
#pragma unroll
    for (int j = 0; j < 8; ++j) {
      int r = orow + j;
      if (r < M) {
        float* Cr = C + (size_t)r * HIDDEN + lm;
        Cr[0]  = acc0[j];
        Cr[16] = acc1[j];
        Cr[32] = acc2[j];
        Cr[48] = acc3[j];
      }
    }
  }
}

// ---------------- edge scatter: out[dst] += h[src] * dinv[src]*dinv[dst] ----------------
// One wave per edge: edge id forced scalar so src/dst/dinv become SMEM loads;
// each lane handles 2 features (b64 load + 2 f32 atomics).
__global__ void k_scatter(const float* __restrict__ h, const int* __restrict__ src,
                          const int* __restrict__ dst, const float* __restrict__ dinv,
                          float* __restrict__ out, int E) {
  int wv = blockIdx.x * (blockDim.x >> 5) + (threadIdx.x >> 5);
  if (wv >= E) return;                       // wave-uniform
  int e = __builtin_amdgcn_readfirstlane(wv);
  int s = src[e];
  int d = dst[e];
  float norm = dinv[s] * dinv[d];
  int f = (threadIdx.x & 31) << 1;           // 2 features per lane
  const float2 hv = *(const float2*)(h + (size_t)s * HIDDEN + f);
  float* o = out + (size_t)d * HIDDEN + f;
  atomicAdd(o,     hv.x * norm);
  atomicAdd(o + 1, hv.y * norm);
}

// ---------------- self-loop + bias + relu (in place on agg) ----------------
__global__ void k_finish(float* __restrict__ agg, const float* __restrict__ hw,
                         const float* __restrict__ dinv, const float* __restrict__ bias, int N) {
  int i = blockIdx.x * blockDim.x + threadIdx.x;
  if (i >= N * HIDDEN) return;
  int n = i >> 6;
  int f = i & 63;
  float di = dinv[n];
  float v = agg[i] + hw[i] * di * di + bias[f];
  agg[i] = fmaxf(v, 0.0f);
}

// ---------------- pooling ----------------
__global__ void k_pool_cnt(const int* __restrict__ batch, float* __restrict__ cnt, int N) {
  int i = blockIdx.x * blockDim.x + threadIdx.x;
  if (i < N) atomicAdd(&cnt[batch[i]], 1.0f);
}

__global__ void k_pool_sum(const float* __restrict__ h, const int* __restrict__ batch,
                           float* __restrict__ sums, int N) {
  int i = blockIdx.x * blockDim.x + threadIdx.x;
  if (i >= N * HIDDEN) return;
  int n = i >> 6;
  int f = i & 63;
  atomicAdd(&sums[(size_t)batch[n] * HIDDEN + f], h[i]);
}

// ---------------- MLP head: out[g] = relu(mean_g @ Wc1 + bc1) @ Wc2 + bc2 ----------------
__global__ void k_head(const float* __restrict__ sums, const float* __restrict__ cnt,
                       const float* __restrict__ Wc1, const float* __restrict__ bc1,
                       const float* __restrict__ Wc2, const float* __restrict__ bc2,
                       float* __restrict__ out, int G) {
  int g = blockIdx.x * blockDim.x + threadIdx.x;
  if (g >= G) return;
  float inv = 1.0f / fmaxf(cnt[g], 1.0f);
  float mean[HIDDEN];
#pragma unroll
  for (int k = 0; k < HIDDEN; ++k) mean[k] = sums[(size_t)g * HIDDEN + k] * inv;
  float r = bc2[0];
  for (int j = 0; j < 32; ++j) {
    float z = bc1[j];
#pragma unroll
    for (int k = 0; k < HIDDEN; ++k) z += mean[k] * Wc1[k * 32 + j];
    z = fmaxf(z, 0.0f);
    r += z * Wc2[j];
  }
  out[g] = r;
}

// ---------------- launcher ----------------
extern "C" void kernel_launch(void* const* d_in, const int* in_sizes, int n_in,
                              void* d_out, int out_size, void* d_ws, size_t ws_size,
                              hipStream_t stream) {
  const float* x    = (const float*)d_in[0];
  const int*   edge = (const int*)d_in[1];
  const int*   batch= (const int*)d_in[2];
  const float* W1   = (const float*)d_in[3];
  const float* b1   = (const float*)d_in[4];
  const float* W2   = (const float*)d_in[5];
  const float* b2   = (const float*)d_in[6];
  const float* Wc1  = (const float*)d_in[7];
  const float* bc1  = (const float*)d_in[8];
  const float* Wc2  = (const float*)d_in[9];
  const float* bc2  = (const float*)d_in[10];
  float* out = (float*)d_out;

  const int N = in_sizes[0] / 128;   // 50000 nodes
  const int E = in_sizes[1] / 2;     // 800000 edges
  const int G = 64;                  // graphs
  const int NH = N * HIDDEN;

  const int* src = edge;
  const int* dst = edge + E;

  float* ws   = (float*)d_ws;
  size_t nPad  = ((size_t)N + 63) & ~(size_t)63;
  size_t nhPad = (((size_t)N + 15) & ~(size_t)15) * HIDDEN;
  float* dinv = ws;
  float* bufA = ws + nPad;
  float* bufB = bufA + nhPad;
  float* psum = bufB + nhPad;        // G*HIDDEN
  float* pcnt = psum + (size_t)G * HIDDEN;

  const int B256 = 256;

  // symmetric normalization: deg = 1 (self-loop) + in-degree; dinv = rsqrt(deg)
  k_fill  <<<(N + B256 - 1) / B256, B256, 0, stream>>>(dinv, 1.0f, N);
  k_degree<<<(E + B256 - 1) / B256, B256, 0, stream>>>(dst, dinv, E);
  k_rsqrt <<<(N + B256 - 1) / B256, B256, 0, stream>>>(dinv, N);

  int tiles = (N + 15) / 16;
  dim3 gblk(128);                       // 4 waves -> 64 rows per block
  dim3 ggrid((tiles + 3) / 4);

  int scatterBlocks = (E + 7) / 8;      // 8 waves (edges) per 256-thread block

  // ---- layer 1 ----
  k_gemm_wmma<128><<<ggrid, gblk, 0, stream>>>(x, W1, bufA, N);
  k_fill   <<<(NH + B256 - 1) / B256, B256, 0, stream>>>(bufB, 0.0f, NH);
  k_scatter<<<scatterBlocks, B256, 0, stream>>>(bufA, src, dst, dinv, bufB, E);
  k_finish <<<(NH + B256 - 1) / B256, B256, 0, stream>>>(bufB, bufA, dinv, b1, N);

  // ---- layer 2 ----
  k_gemm_wmma<64><<<ggrid, gblk, 0, stream>>>(bufB, W2, bufA, N);
  k_fill   <<<(NH + B256 - 1) / B256, B256, 0, stream>>>(bufB, 0.0f, NH);
  k_scatter<<<scatterBlocks, B256, 0, stream>>>(bufA, src, dst, dinv, bufB, E);
  k_finish <<<(NH + B256 - 1) / B256, B256, 0, stream>>>(bufB, bufA, dinv, b2, N);

  // ---- global mean pool + head ----
  k_fill    <<<(G * HIDDEN + G + B256 - 1) / B256, B256, 0, stream>>>(psum, 0.0f, G * HIDDEN + G);
  k_pool_cnt<<<(N + B256 - 1) / B256, B256, 0, stream>>>(batch, pcnt, N);
  k_pool_sum<<<(NH + B256 - 1) / B256, B256, 0, stream>>>(bufB, batch, psum, N);
  k_head    <<<1, G, 0, stream>>>(psum, pcnt, Wc1, bc1, Wc2, bc2, out, G);
}